// FeaturePropagation_68281390072562
// MI455X (gfx1250) — compile-verified
//
#include <hip/hip_runtime.h>
#include <hip/hip_bf16.h>

typedef float v2f __attribute__((ext_vector_type(2)));
typedef float v8f __attribute__((ext_vector_type(8)));

#define BATCH 8
#define NPT1  8192
#define NPT2  2048
#define C1CH  128
#define C2CH  256
#define HCH   256
#define KIN   384          // C2CH + C1CH
#define BN_EPS 1e-5f
#define LPAD  36           // LDS row stride (floats): 16B-aligned rows, conflict-free b64 reads

// ---------------------------------------------------------------------------
// CDNA5 async global->LDS copies (ASYNCcnt path). VDST carries the per-lane
// LDS byte address; low 32 bits of a generic shared pointer are the LDS offset.
// ---------------------------------------------------------------------------
__device__ __forceinline__ void async_copy_b128(unsigned lds_off, const void* gptr) {
    asm volatile("global_load_async_to_lds_b128 %0, %1, off"
                 :: "v"(lds_off), "v"(gptr) : "memory");
}
__device__ __forceinline__ void async_copy_b32(unsigned lds_off, const void* gptr) {
    asm volatile("global_load_async_to_lds_b32 %0, %1, off"
                 :: "v"(lds_off), "v"(gptr) : "memory");
}
__device__ __forceinline__ void wait_async0() {
    asm volatile("s_wait_asynccnt 0x0" ::: "memory");
}
__device__ __forceinline__ unsigned lds_off(const void* p) {
    return (unsigned)(unsigned long long)p;
}

// ---------------------------------------------------------------------------
// Fold BN (eval mode) into conv weights/bias:  W'[h][c] = W[h][c]*s[h],
// b'[h] = (b[h]-m[h])*s[h] + be[h],  s[h] = g[h]*rsqrt(v[h]+eps)
// ---------------------------------------------------------------------------
__global__ __launch_bounds__(256) void fold_bn_kernel(
    const float* __restrict__ w1, const float* __restrict__ b1,
    const float* __restrict__ g1, const float* __restrict__ be1,
    const float* __restrict__ m1, const float* __restrict__ v1,
    const float* __restrict__ w2, const float* __restrict__ b2,
    const float* __restrict__ g2, const float* __restrict__ be2,
    const float* __restrict__ m2, const float* __restrict__ v2,
    float* __restrict__ Wf1, float* __restrict__ bias1,
    float* __restrict__ Wf2, float* __restrict__ bias2)
{
    int t = blockIdx.x * 256 + threadIdx.x;
    if (t < HCH * KIN) {
        int h = t / KIN;
        float s = g1[h] * rsqrtf(v1[h] + BN_EPS);
        Wf1[t] = w1[t] * s;
    }
    if (t < HCH * HCH) {
        int h = t / HCH;
        float s = g2[h] * rsqrtf(v2[h] + BN_EPS);
        Wf2[t] = w2[t] * s;
    }
    if (t < HCH) {
        float s1 = g1[t] * rsqrtf(v1[t] + BN_EPS);
        bias1[t] = (b1[t] - m1[t]) * s1 + be1[t];
        float s2 = g2[t] * rsqrtf(v2[t] + BN_EPS);
        bias2[t] = (b2[t] - m2[t]) * s2 + be2[t];
    }
}

// ---------------------------------------------------------------------------
// kNN (k=3): xyz2 of one batch staged in LDS (24 KB); one lane per query
// point; branch-free sorted-insert keeps the wave converged.
// ---------------------------------------------------------------------------
__global__ __launch_bounds__(256) void knn_kernel(
    const float* __restrict__ xyz1, const float* __restrict__ xyz2,
    int* __restrict__ knn_idx, float* __restrict__ knn_w)
{
    __shared__ float sx[NPT2], sy[NPT2], sz[NPT2];
    const int b = blockIdx.x;
    const float* p2 = xyz2 + (size_t)b * 3 * NPT2;
    for (int i = threadIdx.x; i < NPT2; i += 256) {
        sx[i] = p2[i];
        sy[i] = p2[NPT2 + i];
        sz[i] = p2[2 * NPT2 + i];
    }
    __syncthreads();

    const int n = blockIdx.y * 256 + threadIdx.x;
    const float* p1 = xyz1 + (size_t)b * 3 * NPT1;
    const float qx = p1[n], qy = p1[NPT1 + n], qz = p1[2 * NPT1 + n];

    float d0 = 3.4e38f, d1 = 3.4e38f, d2 = 3.4e38f;
    int   i0 = 0, i1 = 0, i2 = 0;
    for (int j = 0; j < NPT2; ++j) {
        float dx = qx - sx[j], dy = qy - sy[j], dz = qz - sz[j];
        float d = dx * dx + dy * dy + dz * dz;
        bool lt0 = d < d0, lt1 = d < d1, lt2 = d < d2;
        float nd2 = lt1 ? d1 : (lt2 ? d : d2);
        int   ni2 = lt1 ? i1 : (lt2 ? j : i2);
        float nd1 = lt0 ? d0 : (lt1 ? d : d1);
        int   ni1 = lt0 ? i0 : (lt1 ? j : i1);
        float nd0 = lt0 ? d : d0;
        int   ni0 = lt0 ? j : i0;
        d2 = nd2; i2 = ni2; d1 = nd1; i1 = ni1; d0 = nd0; i0 = ni0;
    }
    float w0 = 1.0f / fmaxf(d0, 1e-10f);
    float w1 = 1.0f / fmaxf(d1, 1e-10f);
    float w2 = 1.0f / fmaxf(d2, 1e-10f);
    float inv = 1.0f / (w0 + w1 + w2);
    size_t base = ((size_t)b * NPT1 + n) * 3;
    knn_idx[base + 0] = i0;  knn_w[base + 0] = w0 * inv;
    knn_idx[base + 1] = i1;  knn_w[base + 1] = w1 * inv;
    knn_idx[base + 2] = i2;  knn_w[base + 2] = w2 * inv;
}

// ---------------------------------------------------------------------------
// Inverse-distance interpolation: Ibuf[b][c][n] = sum_k w_k * f2[b][c][idx_k]
// ---------------------------------------------------------------------------
__global__ __launch_bounds__(256) void interp_kernel(
    const float* __restrict__ f2, const int* __restrict__ knn_idx,
    const float* __restrict__ knn_w, float* __restrict__ Ibuf)
{
    size_t e = (size_t)blockIdx.x * 256 + threadIdx.x;   // < B*C2*N1
    int n = (int)(e % NPT1);
    size_t bc = e / NPT1;
    int c = (int)(bc % C2CH);
    int b = (int)(bc / C2CH);
    size_t q = ((size_t)b * NPT1 + n) * 3;
    int   i0 = knn_idx[q], i1 = knn_idx[q + 1], i2 = knn_idx[q + 2];
    float w0 = knn_w[q], w1 = knn_w[q + 1], w2 = knn_w[q + 2];
    const float* row = f2 + ((size_t)b * C2CH + c) * NPT2;
    Ibuf[e] = w0 * row[i0] + w1 * row[i1] + w2 * row[i2];
}

// ---------------------------------------------------------------------------
// Fused GEMM + bias + ReLU using V_WMMA_F32_16X16X4_F32 with double-buffered
// async global->LDS staging (ASYNCcnt). Stage k+1 copies overlap stage-k WMMAs.
//   Y[m][j] = relu( sum_c Wf[m][c] * X[c][j] + bias[m] )
// X column j = b*N1 + n; rows < c1rows from src1, else src2 (fused concat).
// Block: 256 thr = 8 waves as 4(M) x 2(N); wave tile 32x32; block tile 128x64.
// ---------------------------------------------------------------------------
__global__ __launch_bounds__(256) void gemm_bias_relu_kernel(
    const float* __restrict__ Wf, const float* __restrict__ bias,
    const float* __restrict__ src1, int c1rows,
    const float* __restrict__ src2, int c2rows,
    float* __restrict__ out, int Mtot, int Ktot)
{
    __shared__ __align__(16) float Wl[2][128][LPAD];  // [buf][m_local][k_local]
    __shared__ __align__(16) float Xl[2][64][LPAD];   // [buf][n_local][k_local] (transposed)

    const int tid  = threadIdx.x;
    const int lane = tid & 31;
    const int wv   = tid >> 5;
    const int wm   = wv & 3;          // 0..3  (M sub-tile of 32)
    const int wn   = wv >> 2;         // 0..1  (N sub-tile of 32)
    const int m0   = blockIdx.y * 128;
    const int j0   = blockIdx.x * 64;
    const int b    = j0 / NPT1;       // 64 | N1, whole tile in one batch
    const int nb   = j0 % NPT1;

    v8f acc[2][2] = {};

    // issue one stage of async copies into LDS buffer `buf`
    auto load_stage = [&](int kk, int buf) {
        // W tile: 128 x 32, b128 per lane (16B-aligned in both spaces)
        #pragma unroll
        for (int r = 0; r < 4; ++r) {
            int q   = tid + r * 256;
            int row = q >> 3;
            int col = (q & 7) << 2;
            const float* g = Wf + (size_t)(m0 + row) * Ktot + kk + col;
            async_copy_b128(lds_off(&Wl[buf][row][col]), g);
        }
        // X tile: 32 x 64, b32 scattered transposed; concat fused here
        #pragma unroll
        for (int r = 0; r < 8; ++r) {
            int e   = tid + r * 256;
            int row = e >> 6;          // k offset 0..31
            int col = e & 63;          // n offset 0..63
            int c   = kk + row;
            const float* g = (c < c1rows)
                ? src1 + ((size_t)b * c1rows + c) * NPT1 + nb + col
                : src2 + ((size_t)b * c2rows + (c - c1rows)) * NPT1 + nb + col;
            async_copy_b32(lds_off(&Xl[buf][col][row]), g);
        }
    };

    load_stage(0, 0);
    wait_async0();
    __syncthreads();

    for (int kk = 0; kk < Ktot; kk += 32) {
        const int cur = (kk >> 5) & 1;
        if (kk + 32 < Ktot)
            load_stage(kk + 32, cur ^ 1);      // overlaps with compute below

        // --- 8 K-steps of 4 -> 32 WMMAs per wave per stage ---
        #pragma unroll
        for (int k4 = 0; k4 < 32; k4 += 4) {
            const int kl = k4 + 2 * (lane >> 4);   // ISA 16x4 f32 A/B K split
            v2f af[2], bf[2];
            #pragma unroll
            for (int i = 0; i < 2; ++i) {
                int m = wm * 32 + i * 16 + (lane & 15);
                af[i] = *(const v2f*)&Wl[cur][m][kl];
            }
            #pragma unroll
            for (int jt = 0; jt < 2; ++jt) {
                int n = wn * 32 + jt * 16 + (lane & 15);
                bf[jt] = *(const v2f*)&Xl[cur][n][kl];
            }
            #pragma unroll
            for (int i = 0; i < 2; ++i)
                #pragma unroll
                for (int jt = 0; jt < 2; ++jt)
                    acc[i][jt] = __builtin_amdgcn_wmma_f32_16x16x4_f32(
                        false, af[i], false, bf[jt], (short)0, acc[i][jt],
                        false, false);
        }

        wait_async0();       // next-stage copies have landed
        __syncthreads();     // all waves done reading `cur` / writing `cur^1`
    }

    // --- epilogue: bias + ReLU, write [B, Mtot, N1] layout ---
    const int half = lane >> 4;         // C/D layout: VGPR r -> M = r + 8*half
    const int ncol = lane & 15;
    #pragma unroll
    for (int i = 0; i < 2; ++i) {
        #pragma unroll
        for (int jt = 0; jt < 2; ++jt) {
            int n = nb + wn * 32 + jt * 16 + ncol;
            #pragma unroll
            for (int r = 0; r < 8; ++r) {
                int m = m0 + wm * 32 + i * 16 + r + 8 * half;
                float v = acc[i][jt][r] + bias[m];
                v = v > 0.0f ? v : 0.0f;
                out[((size_t)b * Mtot + m) * NPT1 + n] = v;
            }
        }
    }
}

// ---------------------------------------------------------------------------
extern "C" void kernel_launch(void* const* d_in, const int* in_sizes, int n_in,
                              void* d_out, int out_size, void* d_ws, size_t ws_size,
                              hipStream_t stream)
{
    const float* xyz1 = (const float*)d_in[0];
    const float* xyz2 = (const float*)d_in[1];
    const float* f1   = (const float*)d_in[2];
    const float* f2   = (const float*)d_in[3];
    const float* w1   = (const float*)d_in[4];
    const float* b1   = (const float*)d_in[5];
    const float* g1   = (const float*)d_in[6];
    const float* be1  = (const float*)d_in[7];
    const float* m1   = (const float*)d_in[8];
    const float* v1   = (const float*)d_in[9];
    const float* w2   = (const float*)d_in[10];
    const float* b2   = (const float*)d_in[11];
    const float* g2   = (const float*)d_in[12];
    const float* be2  = (const float*)d_in[13];
    const float* m2   = (const float*)d_in[14];
    const float* v2   = (const float*)d_in[15];

    // workspace layout (floats)
    float* ws    = (float*)d_ws;
    float* Ibuf  = ws;                                  // B*C2*N1 = 16,777,216
    float* Ybuf  = Ibuf + (size_t)BATCH * C2CH * NPT1;  // 16,777,216
    float* Wf1   = Ybuf + (size_t)BATCH * HCH * NPT1;   // 98,304
    float* Wf2   = Wf1 + HCH * KIN;                     // 65,536
    float* bias1 = Wf2 + HCH * HCH;                     // 256
    float* bias2 = bias1 + HCH;                         // 256
    float* knnw  = bias2 + HCH;                         // B*N1*3 = 196,608
    int*   knni  = (int*)(knnw + (size_t)BATCH * NPT1 * 3);

    // 1. fold BN into weights/bias
    fold_bn_kernel<<<(HCH * KIN + 255) / 256, 256, 0, stream>>>(
        w1, b1, g1, be1, m1, v1, w2, b2, g2, be2, m2, v2,
        Wf1, bias1, Wf2, bias2);

    // 2. k=3 nearest neighbors + normalized inverse-distance weights
    knn_kernel<<<dim3(BATCH, NPT1 / 256), 256, 0, stream>>>(
        xyz1, xyz2, knni, knnw);

    // 3. feature interpolation into Ibuf [B, C2, N1]
    interp_kernel<<<(unsigned)((size_t)BATCH * C2CH * NPT1 / 256), 256, 0, stream>>>(
        f2, knni, knnw, Ibuf);

    // 4. layer 1: concat(Ibuf, features1) -> Ybuf   (M=256, K=384)
    gemm_bias_relu_kernel<<<dim3(BATCH * NPT1 / 64, HCH / 128), 256, 0, stream>>>(
        Wf1, bias1, Ibuf, C2CH, f1, C1CH, Ybuf, HCH, KIN);

    // 5. layer 2: Ybuf -> d_out   (M=256, K=256)
    gemm_bias_relu_kernel<<<dim3(BATCH * NPT1 / 64, HCH / 128), 256, 0, stream>>>(
        Wf2, bias2, Ybuf, HCH, (const float*)nullptr, 0, (float*)d_out, HCH, HCH);
}